// projection_layer_vm_20091857011276
// MI455X (gfx1250) — compile-verified
//
#include <hip/hip_runtime.h>
#include <stdint.h>

// Problem constants (match reference)
#define N_IN      65536
#define D_DIM     128
#define K_NB      16
#define P_PER_B   32768          // N_OUT * CHILD
#define PTS_TOTAL 65536          // B * P_PER_B

#define WAVES_PER_BLOCK 4
#define PTS_PER_WAVE    4
#define THREADS (WAVES_PER_BLOCK * 32)
#define NBLOCKS (PTS_TOTAL / (WAVES_PER_BLOCK * PTS_PER_WAVE))  // 4096

typedef __attribute__((ext_vector_type(2))) float v2f;
typedef __attribute__((ext_vector_type(8))) float v8f;

__device__ __forceinline__ float lane_bcast_f(float v, int srcLane) {
    return __int_as_float(__builtin_amdgcn_ds_bpermute(srcLane << 2, __float_as_int(v)));
}
__device__ __forceinline__ int lane_bcast_i(int v, int srcLane) {
    return __builtin_amdgcn_ds_bpermute(srcLane << 2, v);
}

__global__ __launch_bounds__(THREADS, 1)
void vmf_projection_wmma_kernel(const float* __restrict__ x,       // [B, N_IN, D]
                                const float* __restrict__ coords,  // [2, N_IN]
                                const int*   __restrict__ idx,     // [B, P, K]
                                const float* __restrict__ oc,      // [2, B*P]
                                const float* __restrict__ sigma_d,
                                const float* __restrict__ kappa_vm,
                                float* __restrict__ out)           // [B*P, D]
{
    // 4 waves x 2 buffers x (16 rows x 128 f32) = 64 KB
    __shared__ float lds[WAVES_PER_BLOCK * 2 * K_NB * D_DIM];

    const int tid  = threadIdx.x;
    const int lane = tid & 31;
    const int wid  = tid >> 5;
    const int kl     = lane & 15;          // column / k-slot within half-wave
    const int khalf2 = (lane >> 4) << 1;   // 0 for lanes 0-15, 2 for lanes 16-31

    const int pt_base = (blockIdx.x * WAVES_PER_BLOCK + wid) * PTS_PER_WAVE;

    const float kap     = kappa_vm[0];
    const float inv_sig = 1.0f / sigma_d[0];

    float* wave_lds = &lds[wid * 2 * K_NB * D_DIM];
    // Low 32 bits of an LDS aperture address = byte offset into LDS (ISA 10.2)
    const uint32_t lds_base = (uint32_t)(uintptr_t)wave_lds;

    // ---- per-point weight/index prep: lane kl handles neighbor kl ----
    auto prep = [&](int pt, int& myidx, float& wl) {
        const int* ip = idx + (size_t)pt * K_NB;   // wave-uniform address -> s_load
        myidx = ip[kl];
        const float nlon  = coords[myidx];
        const float nlat  = coords[N_IN + myidx];
        const float lon_o = oc[pt];
        const float lat_o = oc[PTS_TOTAL + pt];
        const float ct = __sinf(lat_o) * __sinf(nlat)
                       + __cosf(lat_o) * __cosf(nlat) * __cosf(lon_o - nlon);
        wl = __expf(kap * (ct - 1.0f)) * inv_sig;
    };

    // ---- async gather of 16 neighbor rows (512B each) into LDS buffer `sel` ----
    auto issue_gather = [&](int pt, int myidx, int sel) {
        const int b = pt >> 15;                         // pt / P_PER_B
        uint32_t ldsoff = lds_base + (uint32_t)sel * (K_NB * D_DIM * 4)
                        + (uint32_t)lane * 16;
#pragma unroll
        for (int k = 0; k < K_NB; ++k) {
            const int uk = lane_bcast_i(myidx, k);      // idx[k] to all lanes
            const unsigned long long g =
                (unsigned long long)(const char*)x
                + (((size_t)b * N_IN + (uint32_t)uk) * (size_t)(D_DIM * 4))
                + (size_t)lane * 16;
            asm volatile("global_load_async_to_lds_b128 %0, %1, off"
                         :: "v"(ldsoff), "v"(g) : "memory");
            ldsoff += 512;
        }
    };

    // ---- WMMA reduction: out(1x128) = w(1x16) * X(16x128), w replicated over M ----
    auto compute = [&](int pt, float wl, int sel) {
        const float* bufp = wave_lds + sel * (K_NB * D_DIM);

        // A(16x4) f32 layout: lanes0-15 VGPR0/1 = K0/K1 ; lanes16-31 = K2/K3
        v2f a[4];
#pragma unroll
        for (int kt = 0; kt < 4; ++kt) {
            const int kb = kt * 4 + khalf2;
            a[kt].x = lane_bcast_f(wl, kb);
            a[kt].y = lane_bcast_f(wl, kb + 1);
        }

        float* op = out + (size_t)pt * D_DIM;
        // Two adjacent 16-wide d-tiles per iteration; store with all 32 lanes:
        // lanes 0-15 write tile 2dp (row M=0 of cE), lanes 16-31 write tile
        // 2dp+1 (row M=8 of cO == same values, rows identical). Address is
        // op + dp*32 + lane for every lane -> coalesced 128B store, no EXEC
        // divergence between WMMAs.
#pragma unroll
        for (int dp = 0; dp < 4; ++dp) {
            v8f cE = {0.f, 0.f, 0.f, 0.f, 0.f, 0.f, 0.f, 0.f};
            v8f cO = {0.f, 0.f, 0.f, 0.f, 0.f, 0.f, 0.f, 0.f};
            const int dcolE = (dp * 2) * 16 + kl;
            const int dcolO = dcolE + 16;
#pragma unroll
            for (int kt = 0; kt < 4; ++kt) {
                const int kb = kt * 4 + khalf2;
                v2f bmE, bmO;
                bmE.x = bufp[(kb + 0) * D_DIM + dcolE];
                bmE.y = bufp[(kb + 1) * D_DIM + dcolE];
                bmO.x = bufp[(kb + 0) * D_DIM + dcolO];
                bmO.y = bufp[(kb + 1) * D_DIM + dcolO];
                cE = __builtin_amdgcn_wmma_f32_16x16x4_f32(
                        false, a[kt], false, bmE, (short)0, cE, false, false);
                cO = __builtin_amdgcn_wmma_f32_16x16x4_f32(
                        false, a[kt], false, bmO, (short)0, cO, false, false);
            }
            const float val = (lane < 16) ? cE[0] : cO[0];
            op[dp * 32 + lane] = val;
        }
    };

    // ---- software pipeline over PTS_PER_WAVE points, double-buffered ----
    int   midx_cur, midx_nxt;
    float w_cur, w_nxt;
    prep(pt_base, midx_cur, w_cur);
    issue_gather(pt_base, midx_cur, 0);

#pragma unroll
    for (int i = 0; i < PTS_PER_WAVE; ++i) {
        const int sel = i & 1;
        if (i + 1 < PTS_PER_WAVE) {
            prep(pt_base + i + 1, midx_nxt, w_nxt);
            issue_gather(pt_base + i + 1, midx_nxt, sel ^ 1);
            // 16 in flight for next point; wait for current point's 16 to land
            asm volatile("s_wait_asynccnt 16" ::: "memory");
        } else {
            asm volatile("s_wait_asynccnt 0" ::: "memory");
        }
        compute(pt_base + i, w_cur, sel);
        w_cur = w_nxt;
        midx_cur = midx_nxt;
    }
}

extern "C" void kernel_launch(void* const* d_in, const int* in_sizes, int n_in,
                              void* d_out, int out_size, void* d_ws, size_t ws_size,
                              hipStream_t stream) {
    (void)in_sizes; (void)n_in; (void)out_size; (void)d_ws; (void)ws_size;
    const float* x      = (const float*)d_in[0];  // x_level_in  [B, N_IN, D]
    const float* coords = (const float*)d_in[1];  // coords_in   [2, N_IN]
    const int*   idx    = (const int*)  d_in[2];  // indices_in  [B, P, K]
    const float* oc     = (const float*)d_in[3];  // output_coords [2, B*P]
    const float* sig    = (const float*)d_in[4];  // sigma_d [1]
    const float* kap    = (const float*)d_in[5];  // kappa_vm [1]
    float*       out    = (float*)d_out;          // [B*P, D]

    vmf_projection_wmma_kernel<<<NBLOCKS, THREADS, 0, stream>>>(
        x, coords, idx, oc, sig, kap, out);
}